// GCNModel_11897059410619
// MI455X (gfx1250) — compile-verified
//
#include <hip/hip_runtime.h>

#define HD 64

typedef __attribute__((ext_vector_type(2))) float v2f;
typedef __attribute__((ext_vector_type(8))) float v8f;

// ---------- degree / norm (over padded NP rows so GEMM epilogue is guard-free) ----------
__global__ void k_init_deg(float* deg, int NP) {
    int i = blockIdx.x * blockDim.x + threadIdx.x;
    if (i < NP) deg[i] = 1.0f;                 // self-loop contributes 1
}

__global__ void k_deg_edges(const int* __restrict__ ei, float* deg, int E) {
    int e = blockIdx.x * blockDim.x + threadIdx.x;
    if (e < E) atomicAdd(&deg[ei[E + e]], 1.0f);   // dst = ei[1][e]
}

__global__ void k_dinv(float* deg, int NP) {
    int i = blockIdx.x * blockDim.x + threadIdx.x;
    if (i < NP) deg[i] = rsqrtf(deg[i]);
}

// ---------- zero-padding helpers ----------
// x [N,9] -> bufX [NP,12] (cols 9..11 and rows N..NP-1 zero)
__global__ void k_pad_x(const float* __restrict__ x, float* __restrict__ bx, int N, int NP) {
    int i = blockIdx.x * blockDim.x + threadIdx.x;
    if (i >= NP * 12) return;
    int row = i / 12, c = i - row * 12;
    bx[i] = (row < N && c < 9) ? x[row * 9 + c] : 0.0f;
}

// W1 [9,64] -> bufW1 [12,64]
__global__ void k_pad_w(const float* __restrict__ w, float* __restrict__ bw) {
    int i = blockIdx.x * blockDim.x + threadIdx.x;
    if (i >= 12 * HD) return;
    bw[i] = (i < 9 * HD) ? w[i] : 0.0f;
}

__global__ void k_zero(float* p, int n) {
    int i = blockIdx.x * blockDim.x + threadIdx.x;
    if (i < n) p[i] = 0.0f;
}

// ---------- dense transform via f32 WMMA, fused self-loop epilogue ----------
// T[NP,64] = A[NP,K] * W[K,64];  agg = T * dinv^2.  One wave per 16-row tile,
// no per-lane guards anywhere (inputs are padded), EXEC stays all-ones.
template<int K>
__global__ void k_gemm_wmma(const float* __restrict__ A,
                            const float* __restrict__ W,
                            float* __restrict__ T,
                            float* __restrict__ agg,
                            const float* __restrict__ dinv, int tiles) {
    int wave = blockIdx.x * (blockDim.x >> 5) + (threadIdx.x >> 5);
    if (wave >= tiles) return;              // wave-uniform exit
    int lane = threadIdx.x & 31;
    int half = lane >> 4;                   // 0: lanes 0-15, 1: lanes 16-31
    int l15  = lane & 15;
    int row0 = wave * 16;

    v8f acc[4];
#pragma unroll
    for (int j = 0; j < 4; ++j)
#pragma unroll
        for (int r = 0; r < 8; ++r) acc[j][r] = 0.0f;

    const float* Ar = A + (size_t)(row0 + l15) * K;   // A 16x4 frag: this lane's row

#pragma unroll
    for (int k0 = 0; k0 < K; k0 += 4) {
        int kk = k0 + half * 2;             // lanes 0-15: K=k0,k0+1; lanes 16-31: K=k0+2,k0+3
        v2f a;
        a.x = Ar[kk];
        a.y = Ar[kk + 1];
#pragma unroll
        for (int j = 0; j < 4; ++j) {
            int col = j * 16 + l15;
            v2f b;
            b.x = W[(size_t)kk       * HD + col];
            b.y = W[(size_t)(kk + 1) * HD + col];
            acc[j] = __builtin_amdgcn_wmma_f32_16x16x4_f32(
                false, a, false, b, (short)0, acc[j], false, false);
        }
    }

    // this lane's 8 output rows: M = r + half*8
    float d2[8];
#pragma unroll
    for (int r = 0; r < 8; ++r) {
        float d = dinv[row0 + r + half * 8];
        d2[r] = d * d;
    }
#pragma unroll
    for (int j = 0; j < 4; ++j) {
#pragma unroll
        for (int r = 0; r < 8; ++r) {
            size_t idx = (size_t)(row0 + r + half * 8) * HD + j * 16 + l15;
            float v = acc[j][r];
            T[idx]   = v;
            agg[idx] = v * d2[r];
        }
    }
}

// ---------- edge scatter: one wave per edge, lane covers f and f+32 ----------
__global__ void k_scatter(const int* __restrict__ ei, const float* __restrict__ h,
                          const float* __restrict__ dinv, float* __restrict__ agg, int E) {
    int wave = (blockIdx.x * blockDim.x + threadIdx.x) >> 5;
    int lane = threadIdx.x & 31;
    if (wave >= E) return;
    int src = ei[wave];
    int dst = ei[E + wave];
    float w = dinv[src] * dinv[dst];
    const float* hs = h   + (size_t)src * HD;
    float*       ad = agg + (size_t)dst * HD;
    atomicAdd(&ad[lane],      hs[lane]      * w);
    atomicAdd(&ad[lane + 32], hs[lane + 32] * w);
}

// ---------- bias + relu ----------
__global__ void k_relu_bias(const float* __restrict__ agg, const float* __restrict__ b,
                            float* __restrict__ h, int N) {
    int i = blockIdx.x * blockDim.x + threadIdx.x;
    if (i < N * HD) h[i] = fmaxf(agg[i] + b[i & 63], 0.0f);
}

// ---------- pooling ----------
__global__ void k_count(const int* __restrict__ bat, float* counts, int N) {
    int i = blockIdx.x * blockDim.x + threadIdx.x;
    if (i < N) atomicAdd(&counts[bat[i]], 1.0f);
}

__global__ void k_sum(const int* __restrict__ bat, const float* __restrict__ h,
                      float* sums, int N) {
    int i = blockIdx.x * blockDim.x + threadIdx.x;
    if (i < N * HD) atomicAdd(&sums[(size_t)bat[i >> 6] * HD + (i & 63)], h[i]);
}

__global__ void k_emb(const float* __restrict__ sums, const float* __restrict__ counts,
                      float* __restrict__ emb, int B) {
    int i = blockIdx.x * blockDim.x + threadIdx.x;
    if (i < B * HD) emb[i] = sums[i] / fmaxf(counts[i >> 6], 1.0f);
}

__global__ void k_final(const float* __restrict__ emb, const float* __restrict__ Wf,
                        const float* __restrict__ bf, float* __restrict__ out, int B) {
    int b = blockIdx.x * blockDim.x + threadIdx.x;
    if (b >= B) return;
    float s = 0.0f;
#pragma unroll
    for (int f = 0; f < HD; ++f) s += emb[(size_t)b * HD + f] * Wf[f];
    out[b] = s + bf[0];
}

extern "C" void kernel_launch(void* const* d_in, const int* in_sizes, int n_in,
                              void* d_out, int out_size, void* d_ws, size_t ws_size,
                              hipStream_t stream) {
    const float* x   = (const float*)d_in[0];
    const int*   ei  = (const int*)  d_in[1];
    const int*   bat = (const int*)  d_in[2];
    const float* W1  = (const float*)d_in[3];
    const float* b1  = (const float*)d_in[4];
    const float* W2  = (const float*)d_in[5];
    const float* b2  = (const float*)d_in[6];
    const float* W3  = (const float*)d_in[7];
    const float* b3  = (const float*)d_in[8];
    const float* Wf  = (const float*)d_in[9];
    const float* bf  = (const float*)d_in[10];

    const int N  = in_sizes[0] / 9;       // x is [N,9]
    const int E  = in_sizes[1] / 2;       // edge_index is [2,E]
    const int B  = out_size / (1 + HD);   // d_out = out[B,1] ++ emb[B,64]
    const int NP = (N + 15) & ~15;        // rows padded to WMMA tile
    float* out = (float*)d_out;
    float* emb = out + B;

    // workspace layout (floats)
    float* p = (float*)d_ws;
    size_t nh = (size_t)NP * HD;
    float* dinv   = p; p += (size_t)NP;
    float* bufX   = p; p += (size_t)NP * 12;
    float* bufW1  = p; p += 12 * HD;
    float* bufT   = p; p += nh;
    float* bufAgg = p; p += nh;
    float* bufH   = p; p += nh;
    float* sums   = p; p += (size_t)B * HD;
    float* counts = p; p += B;
    (void)ws_size; (void)n_in;

    const int T = 256;
    const int gNP  = (NP + T - 1) / T;
    const int gE   = (E + T - 1) / T;
    const int gNH  = (N * HD + T - 1) / T;
    const int gBH  = (B * HD + T - 1) / T;
    const int gB   = (B + T - 1) / T;
    const int tiles = NP / 16;
    const int gW    = (tiles + 7) / 8;    // 8 waves per 256-thread block
    const int gEw   = (E + 7) / 8;        // 8 edges (one wave each) per block
    const int gX    = (NP * 12 + T - 1) / T;

    // degrees, norm, padded inputs
    k_init_deg <<<gNP, T, 0, stream>>>(dinv, NP);
    k_deg_edges<<<gE,  T, 0, stream>>>(ei, dinv, E);
    k_dinv     <<<gNP, T, 0, stream>>>(dinv, NP);
    k_pad_x    <<<gX,  T, 0, stream>>>(x, bufX, N, NP);
    k_pad_w    <<<3,   T, 0, stream>>>(W1, bufW1);
    if (NP > N) {   // zero padded tail rows of bufH so later-layer GEMMs read zeros
        int tail = (NP - N) * HD;
        k_zero<<<(tail + T - 1) / T, T, 0, stream>>>(bufH + (size_t)N * HD, tail);
    }

    // layer 1 (K = 12, zero-padded)
    k_gemm_wmma<12><<<gW,  T, 0, stream>>>(bufX, bufW1, bufT, bufAgg, dinv, tiles);
    k_scatter      <<<gEw, T, 0, stream>>>(ei, bufT, dinv, bufAgg, E);
    k_relu_bias    <<<gNH, T, 0, stream>>>(bufAgg, b1, bufH, N);

    // layer 2 (K = 64)
    k_gemm_wmma<64><<<gW,  T, 0, stream>>>(bufH, W2, bufT, bufAgg, dinv, tiles);
    k_scatter      <<<gEw, T, 0, stream>>>(ei, bufT, dinv, bufAgg, E);
    k_relu_bias    <<<gNH, T, 0, stream>>>(bufAgg, b2, bufH, N);

    // layer 3 (K = 64)
    k_gemm_wmma<64><<<gW,  T, 0, stream>>>(bufH, W3, bufT, bufAgg, dinv, tiles);
    k_scatter      <<<gEw, T, 0, stream>>>(ei, bufT, dinv, bufAgg, E);
    k_relu_bias    <<<gNH, T, 0, stream>>>(bufAgg, b3, bufH, N);

    // global mean pool + head
    k_zero <<<gBH, T, 0, stream>>>(sums, B * HD);
    k_zero <<<gB,  T, 0, stream>>>(counts, B);
    k_count<<<gNP > 0 ? (N + T - 1) / T : 1, T, 0, stream>>>(bat, counts, N);
    k_sum  <<<gNH, T, 0, stream>>>(bat, bufH, sums, N);
    k_emb  <<<gBH, T, 0, stream>>>(sums, counts, emb, B);
    k_final<<<gB,  T, 0, stream>>>(emb, Wf, bf, out, B);
}